// LSTM_54107997995127
// MI455X (gfx1250) — compile-verified
//
#include <hip/hip_runtime.h>

// LSTM (INPUT=1, HIDDEN=2, SEQ=4) + FC(2->1) over BATCH elements.
// Strategy: per wave32, tiles of 16 batch elements; each step's 8-gate
// pre-activation computed with V_WMMA_F32_16X16X4_F32:
//   A (16x4, const)  : row g = [W_ih[g], W_hh[g,0], W_hh[g,1], b_ih[g]+b_hh[g]]
//   B (4x16)         : col n = [x_t, h0, h1, 1]^T for batch base+n
//   D (16x16)        : lane n (<16), vgpr r -> gate r of batch n (all in-lane)
// Elementwise LSTM update runs on the TRANS pipe (v_exp/v_rcp).
// EXEC stays all-ones through every WMMA: OOB handling is done by clamping
// load addresses (arithmetic min) and guarding only the final store.

typedef __attribute__((ext_vector_type(2))) float v2f;
typedef __attribute__((ext_vector_type(4))) float v4f;
typedef __attribute__((ext_vector_type(8))) float v8f;

__device__ __forceinline__ float fast_sigmoid(float x) {
    // 1 / (1 + exp(-x)) : v_exp_f32 + v_rcp_f32 (TRANS pipe)
    return __builtin_amdgcn_rcpf(1.0f + __expf(-x));
}
__device__ __forceinline__ float fast_tanh(float x) {
    // tanh(x) = 2*sigmoid(2x) - 1
    return fmaf(2.0f, fast_sigmoid(2.0f * x), -1.0f);
}

#define TILES_PER_WAVE 8

__global__ __launch_bounds__(256) void lstm_wmma_kernel(
    const float* __restrict__ x,     // [B, 4, 1]
    const float* __restrict__ W_ih,  // [8, 1]
    const float* __restrict__ W_hh,  // [8, 2] row-major
    const float* __restrict__ b_ih,  // [8]
    const float* __restrict__ b_hh,  // [8]
    const float* __restrict__ W_fc,  // [1, 2]
    const float* __restrict__ b_fc,  // [1]
    float* __restrict__ out,         // [B]
    int batch)
{
    const int tid   = blockIdx.x * blockDim.x + threadIdx.x;
    const int wave  = tid >> 5;
    const int lane  = threadIdx.x & 31;
    const int n     = lane & 15;            // batch column within tile
    const bool lower = (lane < 16);

    // ---- Build constant A operand (gate-major weights), per ISA A layout:
    // lane m<16 holds A[m][0],A[m][1]; lane m+16 holds A[m][2],A[m][3].
    v2f a;
    if (n < 8) {
        if (lower) { a.x = W_ih[n];          a.y = W_hh[2 * n]; }
        else       { a.x = W_hh[2 * n + 1];  a.y = b_ih[n] + b_hh[n]; }
    } else {
        a.x = 0.0f; a.y = 0.0f;
    }

    const float wfc0 = W_fc[0];
    const float wfc1 = W_fc[1];
    const float bfc  = b_fc[0];

    const long base0 = (long)wave * (TILES_PER_WAVE * 16);
    const v4f* xp = reinterpret_cast<const v4f*>(x);

    for (int t = 0; t < TILES_PER_WAVE; ++t) {
        const long base = base0 + (long)t * 16;   // wave-uniform
        if (base >= batch) break;                 // wave-uniform exit

        // Streaming load of this element's 4 timesteps (16B, non-temporal).
        // Clamp (no EXEC predication) so the load is unconditional.
        const long bi  = base + n;                // lanes 16..31 mirror 0..15
        const long lbi = bi < (long)batch ? bi : (long)batch - 1;
        v4f xv = __builtin_nontemporal_load(xp + lbi);
        // Speculative prefetch of the next tile (dropped silently if OOB).
        __builtin_prefetch((const void*)(xp + lbi + 16), 0, 0);
        float xt[4] = {xv.x, xv.y, xv.z, xv.w};

        float h0 = 0.0f, h1 = 0.0f, c0 = 0.0f, c1 = 0.0f;

#pragma unroll
        for (int s = 0; s < 4; ++s) {
            // B operand, per ISA B layout: lower lanes hold K=0,1; upper K=2,3.
            const float h1_sw = __shfl_xor(h1, 16, 32);  // h1[n] -> lane n+16
            v2f b;
            b.x = lower ? xt[s] : h1_sw;   // K=0: x_t      | K=2: h1
            b.y = lower ? h0    : 1.0f;    // K=1: h0       | K=3: 1 (bias)

            v8f d = {};
            d = __builtin_amdgcn_wmma_f32_16x16x4_f32(
                    /*neg_a=*/false, a, /*neg_b=*/false, b,
                    /*c_mod=*/(short)0, d, /*reuse_a=*/false, /*reuse_b=*/false);

            // Lane n (<16) holds gates [i0,i1,f0,f1,g0,g1,o0,o1] of batch base+n.
            const float ig0 = fast_sigmoid(d[0]);
            const float ig1 = fast_sigmoid(d[1]);
            const float fg0 = fast_sigmoid(d[2]);
            const float fg1 = fast_sigmoid(d[3]);
            const float gg0 = fast_tanh(d[4]);
            const float gg1 = fast_tanh(d[5]);
            const float og0 = fast_sigmoid(d[6]);
            const float og1 = fast_sigmoid(d[7]);

            c0 = fmaf(fg0, c0, ig0 * gg0);
            c1 = fmaf(fg1, c1, ig1 * gg1);
            h0 = og0 * fast_tanh(c0);
            h1 = og1 * fast_tanh(c1);
        }

        // FC + store (lanes 0..15 hold the real results). Divergence here is
        // after all WMMAs of this iteration; reconverges before the next one.
        if (lower && bi < (long)batch) {
            const float r = fmaf(wfc1, h1, fmaf(wfc0, h0, bfc));
            __builtin_nontemporal_store(r, out + bi);
        }
    }
}

extern "C" void kernel_launch(void* const* d_in, const int* in_sizes, int n_in,
                              void* d_out, int out_size, void* d_ws, size_t ws_size,
                              hipStream_t stream) {
    const float* x    = (const float*)d_in[0];
    const float* W_ih = (const float*)d_in[1];
    const float* W_hh = (const float*)d_in[2];
    const float* b_ih = (const float*)d_in[3];
    const float* b_hh = (const float*)d_in[4];
    const float* W_fc = (const float*)d_in[5];
    const float* b_fc = (const float*)d_in[6];
    float* out = (float*)d_out;

    const int batch = in_sizes[0] / 4;                  // x is [B, 4, 1]
    const int tiles = (batch + 15) / 16;
    const int waves = (tiles + TILES_PER_WAVE - 1) / TILES_PER_WAVE;
    const int threads = waves * 32;
    const int block = 256;
    const int grid = (threads + block - 1) / block;

    lstm_wmma_kernel<<<grid, block, 0, stream>>>(
        x, W_ih, W_hh, b_ih, b_hh, W_fc, b_fc, out, batch);
}